// CoAttentionLayerDrugBank_47081431499268
// MI455X (gfx1250) — compile-verified
//
#include <hip/hip_runtime.h>
#include <hip/hip_bf16.h>
#include <math.h>

typedef __attribute__((ext_vector_type(2))) float v2f;
typedef __attribute__((ext_vector_type(8))) float v8f;

#define NFEAT 128
#define NSEG  4096

// ---------------------------------------------------------------------------
// Kernel 1: Y[nrows x 128] = X[nrows x 128] @ W[128 x 128], full f32 via
// v_wmma_f32_16x16x4_f32.  Block = 256 threads = 8 waves; block handles a
// 16-row strip, wave w handles column tile [16w, 16w+16).
// ---------------------------------------------------------------------------
__global__ void proj_gemm_wmma(const float* __restrict__ X,
                               const float* __restrict__ W,
                               float* __restrict__ Y) {
  const int wave = threadIdx.x >> 5;
  const int lane = threadIdx.x & 31;
  const int m0   = blockIdx.x * 16;
  const int n0   = wave * 16;

  const int idx16 = lane & 15;   // M for A, N for B/D
  const int half  = lane >> 4;   // K-half for A/B, row-half for D

  const float* __restrict__ Arow = X + (size_t)(m0 + idx16) * NFEAT;

  v8f c = {};
  #pragma unroll 4
  for (int k0 = 0; k0 < NFEAT; k0 += 4) {
    const int ka = k0 + 2 * half;
    // A(m=idx16, k=ka..ka+1): contiguous -> b64 load
    v2f a;
    a.x = Arow[ka];
    a.y = Arow[ka + 1];
    // B(k=ka..ka+1, n=idx16)
    v2f b;
    b.x = W[(size_t)ka * NFEAT + n0 + idx16];
    b.y = W[(size_t)(ka + 1) * NFEAT + n0 + idx16];
    c = __builtin_amdgcn_wmma_f32_16x16x4_f32(
        /*neg_a=*/false, a, /*neg_b=*/false, b,
        /*c_mod=*/(short)0, c, /*reuse_a=*/false, /*reuse_b=*/false);
  }

  // D(m = v + 8*half, n = idx16)
  float* __restrict__ Yt = Y + (size_t)m0 * NFEAT + n0;
  #pragma unroll
  for (int v = 0; v < 8; ++v) {
    Yt[(size_t)(v + 8 * half) * NFEAT + idx16] = c[v];
  }
}

// ---------------------------------------------------------------------------
// Kernel 2: per-edge  alpha[e] = sum_f PReLU(xj[src][f]+xi[dst][f]+bias[f]) * mlpW[f] + mlp_b
// One wave32 per edge; 4 feats per lane via float4; shuffle reduction.
// ---------------------------------------------------------------------------
__global__ void edge_alpha_kernel(const float* __restrict__ xj,
                                  const float* __restrict__ xi,
                                  const int* __restrict__ src,
                                  const int* __restrict__ dst,
                                  const float* __restrict__ bias,
                                  const float* __restrict__ mlpW,
                                  const float* __restrict__ prelu_w,
                                  const float* __restrict__ mlp_b,
                                  float* __restrict__ alpha,
                                  int E) {
  const int lane   = threadIdx.x & 31;
  const int gwave  = (int)((blockIdx.x * blockDim.x + threadIdx.x) >> 5);
  const int nwaves = (int)((gridDim.x * blockDim.x) >> 5);

  const float slope = prelu_w[0];
  const float bterm = mlp_b[0];
  const float4 bb = ((const float4*)bias)[lane];
  const float4 ww = ((const float4*)mlpW)[lane];

  for (int e = gwave; e < E; e += nwaves) {
    const int s = src[e];
    const int d = dst[e];
    const float4 aj = ((const float4*)(xj + (size_t)s * NFEAT))[lane];
    const float4 ai = ((const float4*)(xi + (size_t)d * NFEAT))[lane];

    float hx = aj.x + ai.x + bb.x;
    float hy = aj.y + ai.y + bb.y;
    float hz = aj.z + ai.z + bb.z;
    float hw = aj.w + ai.w + bb.w;
    hx = hx >= 0.0f ? hx : slope * hx;
    hy = hy >= 0.0f ? hy : slope * hy;
    hz = hz >= 0.0f ? hz : slope * hz;
    hw = hw >= 0.0f ? hw : slope * hw;

    float p = hx * ww.x + hy * ww.y + hz * ww.z + hw * ww.w;
    #pragma unroll
    for (int off = 16; off > 0; off >>= 1)
      p += __shfl_xor(p, off, 32);

    if (lane == 0) alpha[e] = p + bterm;
  }
}

// ---------------------------------------------------------------------------
// Segmented softmax helpers: monotonic uint encoding for float atomicMax.
// ---------------------------------------------------------------------------
__device__ __forceinline__ unsigned fkey(float x) {
  unsigned u = __float_as_uint(x);
  return (u & 0x80000000u) ? ~u : (u | 0x80000000u);
}
__device__ __forceinline__ float funkey(unsigned k) {
  unsigned u = (k & 0x80000000u) ? (k & 0x7fffffffu) : ~k;
  return __uint_as_float(u);
}

__global__ void seg_init_kernel(unsigned* __restrict__ segmax,
                                float* __restrict__ segsum) {
  const int i = blockIdx.x * blockDim.x + threadIdx.x;
  if (i < NSEG) {
    segmax[i] = fkey(-INFINITY);
    segsum[i] = 0.0f;
  }
}

__global__ void seg_max_kernel(const float* __restrict__ alpha,
                               const int* __restrict__ batch,
                               unsigned* __restrict__ segmax, int E) {
  const int stride = gridDim.x * blockDim.x;
  for (int i = blockIdx.x * blockDim.x + threadIdx.x; i < E; i += stride) {
    atomicMax(&segmax[batch[i]], fkey(alpha[i]));
  }
}

__global__ void seg_exp_sum_kernel(const float* __restrict__ alpha,
                                   const int* __restrict__ batch,
                                   const unsigned* __restrict__ segmax,
                                   float* __restrict__ segsum,
                                   float* __restrict__ out, int E) {
  const int stride = gridDim.x * blockDim.x;
  for (int i = blockIdx.x * blockDim.x + threadIdx.x; i < E; i += stride) {
    const int b = batch[i];
    const float ev = expf(alpha[i] - funkey(segmax[b]));
    out[i] = ev;
    atomicAdd(&segsum[b], ev);
  }
}

__global__ void seg_div_kernel(const int* __restrict__ batch,
                               const float* __restrict__ segsum,
                               float* __restrict__ out, int E) {
  const int stride = gridDim.x * blockDim.x;
  for (int i = blockIdx.x * blockDim.x + threadIdx.x; i < E; i += stride) {
    out[i] = out[i] / (segsum[batch[i]] + 1e-16f);
  }
}

// ---------------------------------------------------------------------------
// Launch.  Input order: x_j, x_i, edge_index, edge_index_batch, w_j, w_i,
//                       bias, prelu_w, mlp_W, mlp_b
// ---------------------------------------------------------------------------
extern "C" void kernel_launch(void* const* d_in, const int* in_sizes, int n_in,
                              void* d_out, int out_size, void* d_ws, size_t ws_size,
                              hipStream_t stream) {
  const float* x_j    = (const float*)d_in[0];
  const float* x_i    = (const float*)d_in[1];
  const int*   eidx   = (const int*)d_in[2];     // [2, E] row-major
  const int*   batch  = (const int*)d_in[3];     // [E]
  const float* w_j    = (const float*)d_in[4];
  const float* w_i    = (const float*)d_in[5];
  const float* bias   = (const float*)d_in[6];
  const float* prelu  = (const float*)d_in[7];
  const float* mlp_W  = (const float*)d_in[8];
  const float* mlp_b  = (const float*)d_in[9];

  const int N = in_sizes[0] / NFEAT;   // 100000 nodes
  const int E = in_sizes[3];           // 1000000 edges

  float* ws = (float*)d_ws;
  float*    xj_p   = ws;                                  // N*128
  float*    xi_p   = ws + (size_t)N * NFEAT;              // N*128
  float*    alpha  = ws + 2 * (size_t)N * NFEAT;          // E
  unsigned* segmax = (unsigned*)(alpha + E);              // NSEG
  float*    segsum = (float*)(segmax + NSEG);             // NSEG

  const int* src = eidx;
  const int* dst = eidx + E;
  float* out = (float*)d_out;

  // 1) projections via WMMA (N divisible by 16: 100000/16 = 6250)
  const int row_tiles = N / 16;
  proj_gemm_wmma<<<row_tiles, 256, 0, stream>>>(x_j, w_j, xj_p);
  proj_gemm_wmma<<<row_tiles, 256, 0, stream>>>(x_i, w_i, xi_p);

  // 2) per-edge alpha (wave32 per edge)
  edge_alpha_kernel<<<2048, 256, 0, stream>>>(xj_p, xi_p, src, dst, bias,
                                              mlp_W, prelu, mlp_b, alpha, E);

  // 3) segmented softmax
  seg_init_kernel<<<(NSEG + 255) / 256, 256, 0, stream>>>(segmax, segsum);
  seg_max_kernel<<<2048, 256, 0, stream>>>(alpha, batch, segmax, E);
  seg_exp_sum_kernel<<<2048, 256, 0, stream>>>(alpha, batch, segmax, segsum, out, E);
  seg_div_kernel<<<2048, 256, 0, stream>>>(batch, segsum, out, E);
}